// NPairLoss_78099685310610
// MI455X (gfx1250) — compile-verified
//
#include <hip/hip_runtime.h>
#include <math.h>

#define BN   8192
#define DK   128
#define NEG_BIG -3.0e38f

typedef __attribute__((ext_vector_type(2))) float v2f;
typedef __attribute__((ext_vector_type(8))) float v8f;
typedef __attribute__((address_space(3))) float lds_f;

// ---------------------------------------------------------------- threefry
__device__ __forceinline__ unsigned rotl32(unsigned x, int r) {
    return (x << r) | (x >> (32 - r));
}

// Reproduces jax.random.gumbel(jax.random.key(42), (8192,8192), f32)[i,j]
__device__ float gumbel_f(unsigned i, unsigned j) {
    const unsigned k0 = 0u, k1 = 42u;
    const unsigned k2 = k0 ^ k1 ^ 0x1BD11BDAu;
    unsigned f = i * (unsigned)BN + j;            // < 2^26
    const unsigned half = ((unsigned)BN * (unsigned)BN) >> 1;
    unsigned p  = (f < half) ? f : f - half;
    unsigned x0 = p, x1 = p + half;
    x0 += k0; x1 += k1;
#define TFR(r) { x0 += x1; x1 = rotl32(x1, r); x1 ^= x0; }
    TFR(13) TFR(15) TFR(26) TFR(6)
    x0 += k1; x1 += k2 + 1u;
    TFR(17) TFR(29) TFR(16) TFR(24)
    x0 += k2; x1 += k0 + 2u;
    TFR(13) TFR(15) TFR(26) TFR(6)
    x0 += k0; x1 += k1 + 3u;
    TFR(17) TFR(29) TFR(16) TFR(24)
    x0 += k1; x1 += k2 + 4u;
    TFR(13) TFR(15) TFR(26) TFR(6)
    x0 += k2; x1 += k0 + 5u;
#undef TFR
    unsigned bits = (f < half) ? x0 : x1;
    unsigned fb = (bits >> 9) | 0x3f800000u;
    float u = __uint_as_float(fb) - 1.0f;         // [0,1)
    const float tiny = 1.1754944e-38f;
    u = u * (1.0f - tiny) + tiny;
    u = fmaxf(u, tiny);
    return -__logf(-__logf(u));
}

// ---------------------------------------------------------------- init
__global__ void init_k(int* hist) {
    if (threadIdx.x < 64) hist[threadIdx.x] = 0;
}

// ---------------------------------------------------------------- sq norms
__global__ __launch_bounds__(256) void sq_k(const float* __restrict__ E,
                                            float* __restrict__ sq) {
    int wave = threadIdx.x >> 5, lane = threadIdx.x & 31;
    int row = blockIdx.x * 8 + wave;
    float4 a = ((const float4*)(E + row * DK))[lane];
    float p = a.x * a.x + a.y * a.y + a.z * a.z + a.w * a.w;
    for (int m = 16; m >= 1; m >>= 1) p += __shfl_xor(p, m, 32);
    if (lane == 0) sq[row] = p;
}

// ---------------------------------------------------------------- histogram
__global__ void hist_k(const int* __restrict__ lab, int* hist) {
    int i = blockIdx.x * blockDim.x + threadIdx.x;
    if (i < BN) atomicAdd(&hist[lab[i]], 1);
}

// ---------------------------------------------------------------- gumbel argmax over positives
__global__ __launch_bounds__(256) void gum_k(const int* __restrict__ lab,
                                             int* __restrict__ jidx) {
    __shared__ int   slab[BN];
    __shared__ float bg[256];
    __shared__ int   bj[256];
    int t = threadIdx.x;
    for (int e = t; e < BN / 4; e += 256)
        ((int4*)slab)[e] = ((const int4*)lab)[e];
    __syncthreads();
    int row = blockIdx.x;
    int myl = slab[row];
    float bestg = NEG_BIG; int bestj = 0x7FFFFFFF;
    for (int j = t; j < BN; j += 256) {
        if (j != row && slab[j] == myl) {
            float g = gumbel_f((unsigned)row, (unsigned)j);
            if (g > bestg) { bestg = g; bestj = j; }
        }
    }
    bg[t] = bestg; bj[t] = bestj;
    __syncthreads();
    for (int step = 128; step >= 1; step >>= 1) {
        if (t < step) {
            float g2 = bg[t + step]; int j2 = bj[t + step];
            if (g2 > bg[t] || (g2 == bg[t] && j2 < bj[t])) { bg[t] = g2; bj[t] = j2; }
        }
        __syncthreads();
    }
    if (t == 0) jidx[row] = (bj[0] == 0x7FFFFFFF) ? -1 : bj[0];
}

// ---------------------------------------------------------------- async global->LDS stage-in
// Issues this thread's 8 GLOBAL_LOAD_ASYNC_TO_LDS_B128 transfers for one
// 64-column block (64 rows x 128 K f32, LDS row stride 68 floats).
__device__ __forceinline__ void issue_block(const float* __restrict__ E, int c,
                                            unsigned lds_base, int t) {
#pragma unroll
    for (int q = 0; q < 8; ++q) {
        int e   = q * 256 + t;              // float4 index, 0..2047
        int row = e >> 5;
        int kp  = e & 31;
        unsigned loff = lds_base + (unsigned)((row * 68 + kp * 4) * 4);
        const float* g = E + (c * 64 + row) * DK + kp * 4;
        asm volatile("global_load_async_to_lds_b128 %0, %1, off"
                     :: "v"(loff), "v"(g) : "memory");
    }
}

// ---------------------------------------------------------------- fused WMMA GEMM + online LSE
// grid (64, 8): x = 128-row strip, y = column split (1024 cols each)
__global__ __launch_bounds__(256) void main_lse(const float* __restrict__ E,
                                                const int*   __restrict__ lab,
                                                const float* __restrict__ sq,
                                                float* __restrict__ pm,
                                                float* __restrict__ ps) {
    __shared__ __align__(16) float ldsB[2][64 * 68];  // double-buffered column block
    const int t    = threadIdx.x;
    const int lane = t & 31;
    const int w    = t >> 5;
    const int r0   = blockIdx.x * 128;
    const int cs   = blockIdx.y;              // 0..7
    const int n    = lane & 15;
    const int koff = (lane >= 16) ? 2 : 0;    // 32-bit A/B frag: half-waves own K pairs {0,1}/{2,3}
    const int hrow = (lane >= 16) ? 8 : 0;    // C layout: VGPR r -> M=r (lo half) / r+8 (hi half)

    unsigned base0 = (unsigned)(unsigned long long)(lds_f*)&ldsB[0][0];
    unsigned base1 = (unsigned)(unsigned long long)(lds_f*)&ldsB[1][0];

    const int ibase = r0 + w * 16;
    float sqi[8]; int labi[8];
    for (int r = 0; r < 8; ++r) {
        int ir = ibase + r + hrow;
        sqi[r] = sq[ir]; labi[r] = lab[ir];
    }
    float m[8], s[8];
    for (int r = 0; r < 8; ++r) { m[r] = NEG_BIG; s[r] = 0.0f; }

    const int c0 = cs * 16, cend = cs * 16 + 16;
    issue_block(E, c0, base0, t);             // prologue: async fill buffer 0

    // A fragments are invariant across all column blocks: load once into regs.
    const float* Arow = E + (ibase + n) * DK; // this lane's A-matrix row
    v2f afrag[32];
#pragma unroll
    for (int ks = 0; ks < 32; ++ks)
        afrag[ks] = *(const v2f*)(Arow + ks * 4 + koff);

    for (int c = c0; c < cend; ++c) {
        if (c + 1 < cend) {
            issue_block(E, c + 1, ((c + 1) & 1) ? base1 : base0, t);
            asm volatile("s_wait_asynccnt 0x8" ::: "memory");  // prev block's 8 done
        } else {
            asm volatile("s_wait_asynccnt 0x0" ::: "memory");
        }
        __syncthreads();                      // all waves' current buffer complete

        const float* bufc = (c & 1) ? &ldsB[1][0] : &ldsB[0][0];
        v8f acc[4] = {};
#pragma unroll
        for (int ks = 0; ks < 32; ++ks) {
            int kb = ks * 4 + koff;
#pragma unroll
            for (int nt = 0; nt < 4; ++nt) {
                v2f b = *(const v2f*)&bufc[(nt * 16 + n) * 68 + kb];
                acc[nt] = __builtin_amdgcn_wmma_f32_16x16x4_f32(
                    false, afrag[ks], false, b, (short)0, acc[nt], false, false);
            }
        }

        // online logsumexp update over negatives
#pragma unroll
        for (int nt = 0; nt < 4; ++nt) {
            int j = c * 64 + nt * 16 + n;
            float sqj = sq[j];
            int   lj  = lab[j];
#pragma unroll
            for (int r = 0; r < 8; ++r) {
                if (lj != labi[r]) {                 // neg_mask (diag auto-excluded)
                    float d2 = fmaxf(sqi[r] + sqj - 2.0f * acc[nt][r], 0.0f);
                    float sv = -10.0f * d2;          // -d2 / 0.1
                    float mn = fmaxf(m[r], sv);
                    s[r] = s[r] * __expf(m[r] - mn) + __expf(sv - mn);
                    m[r] = mn;
                }
            }
        }
        __syncthreads();                      // all waves done reading buffer
    }

    // reduce (m,s) across the 16 lanes of each half-wave
    for (int mask = 1; mask < 16; mask <<= 1) {
#pragma unroll
        for (int r = 0; r < 8; ++r) {
            float m2 = __shfl_xor(m[r], mask, 32);
            float s2 = __shfl_xor(s[r], mask, 32);
            float mn = fmaxf(m[r], m2);
            s[r] = s[r] * __expf(m[r] - mn) + s2 * __expf(m2 - mn);
            m[r] = mn;
        }
    }
    if (n == 0) {   // lanes 0 and 16 write rows r / r+8
        for (int r = 0; r < 8; ++r) {
            int row = ibase + r + hrow;
            pm[cs * BN + row] = m[r];
            ps[cs * BN + row] = s[r];
        }
    }
}

// ---------------------------------------------------------------- merge column splits
__global__ void merge_k(const float* __restrict__ pm, const float* __restrict__ ps,
                        float* __restrict__ denom) {
    int i = blockIdx.x * 256 + threadIdx.x;
    float m = NEG_BIG, s = 0.0f;
    for (int c = 0; c < 8; ++c) {
        float m2 = pm[c * BN + i], s2 = ps[c * BN + i];
        float mn = fmaxf(m, m2);
        s = s * __expf(m - mn) + s2 * __expf(m2 - mn);
        m = mn;
    }
    denom[i] = m + __logf(s);
}

// ---------------------------------------------------------------- numerator
__global__ __launch_bounds__(256) void num_k(const float* __restrict__ E,
                                             const float* __restrict__ sq,
                                             const int* __restrict__ jidx,
                                             float* __restrict__ num) {
    int wave = threadIdx.x >> 5, lane = threadIdx.x & 31;
    int row = blockIdx.x * 8 + wave;
    int j = jidx[row];
    if (j < 0) { if (lane == 0) num[row] = 0.0f; return; }
    float4 a = ((const float4*)(E + row * DK))[lane];
    float4 b = ((const float4*)(E + j   * DK))[lane];
    float p = a.x * b.x + a.y * b.y + a.z * b.z + a.w * b.w;
    for (int m = 16; m >= 1; m >>= 1) p += __shfl_xor(p, m, 32);
    if (lane == 0) {
        float d2 = fmaxf(sq[row] + sq[j] - 2.0f * p, 0.0f);
        num[row] = -10.0f * d2;
    }
}

// ---------------------------------------------------------------- final masked mean
__global__ __launch_bounds__(256) void final_k(const int* __restrict__ lab,
                                               const int* __restrict__ hist,
                                               const float* __restrict__ denom,
                                               const float* __restrict__ num,
                                               float* __restrict__ out) {
    __shared__ float red[256];
    float acc = 0.0f;
    for (int i = threadIdx.x; i < BN; i += 256) {
        int c = hist[lab[i]];
        bool valid = (c >= 2) && (c < BN);   // has positive & has negative
        if (valid) acc += denom[i] - num[i];
    }
    red[threadIdx.x] = acc;
    __syncthreads();
    for (int step = 128; step >= 1; step >>= 1) {
        if (threadIdx.x < step) red[threadIdx.x] += red[threadIdx.x + step];
        __syncthreads();
    }
    if (threadIdx.x == 0) out[0] = red[0] / (float)BN;
}

// ---------------------------------------------------------------- launch
extern "C" void kernel_launch(void* const* d_in, const int* in_sizes, int n_in,
                              void* d_out, int out_size, void* d_ws, size_t ws_size,
                              hipStream_t stream) {
    const float* E   = (const float*)d_in[0];
    const int*   lab = (const int*)d_in[1];
    float* out = (float*)d_out;

    // workspace layout
    float* sq    = (float*)d_ws;          // 8192
    float* num   = sq + BN;               // 8192
    float* denom = num + BN;              // 8192
    float* pm    = denom + BN;            // 8*8192
    float* ps    = pm + 8 * BN;           // 8*8192
    int*   jidx  = (int*)(ps + 8 * BN);   // 8192
    int*   hist  = jidx + BN;             // 64

    init_k<<<1, 64, 0, stream>>>(hist);
    sq_k<<<BN / 8, 256, 0, stream>>>(E, sq);
    hist_k<<<BN / 256, 256, 0, stream>>>(lab, hist);
    gum_k<<<BN, 256, 0, stream>>>(lab, jidx);
    main_lse<<<dim3(64, 8), 256, 0, stream>>>(E, lab, sq, pm, ps);
    merge_k<<<BN / 256, 256, 0, stream>>>(pm, ps, denom);
    num_k<<<BN / 8, 256, 0, stream>>>(E, sq, jidx, num);
    final_k<<<1, 256, 0, stream>>>(lab, hist, denom, num, out);
}